// KernelCNP_73306501808774
// MI455X (gfx1250) — compile-verified
//
#include <hip/hip_runtime.h>
#include <hip/hip_bf16.h>

// ---- problem constants (match reference) ----
#define BATCH     8
#define NPTS      2048
#define MPTS      2048
#define NGRID     544          // ceil(128*4.2/16)*16
#define RHO_C     8
#define SIGMA_CH  1024
#define XMIN_F    (-2.1f)
#define GRID_STEP (4.2f / 543.0f)   // (XMAX-XMIN)/(NGRID-1)

typedef __attribute__((ext_vector_type(2))) float v2f;
typedef __attribute__((ext_vector_type(8))) float v8f;

// ------------------------------------------------------------------
// Kernel 1: F[b,g,{den,conv}] via RBF sums over N context points,
// then tiny 2->8 sigmoid MLP + 8x8 rho matmul -> h[b,g,0:8].
// One 256-thread block per (b,g).
// ------------------------------------------------------------------
__global__ __launch_bounds__(256)
void knp_encode(const float* __restrict__ x, const float* __restrict__ y,
                const float* __restrict__ enc_sigma, const float* __restrict__ enc_W,
                const float* __restrict__ enc_b, const float* __restrict__ rho_W,
                const float* __restrict__ rho_b, float* __restrict__ h_out) {
  __shared__ float s0[256];
  __shared__ float s1[256];
  const int bg = blockIdx.x;           // 0 .. BATCH*NGRID-1
  const int b  = bg / NGRID;
  const int g  = bg % NGRID;
  const float gx = XMIN_F + (float)g * GRID_STEP;
  const float k0 = -0.5f * __expf(-2.0f * enc_sigma[0]);
  const float k1 = -0.5f * __expf(-2.0f * enc_sigma[1]);
  const float* xb = x + b * NPTS;
  const float* yb = y + b * NPTS;

  float a0 = 0.0f, a1 = 0.0f;
  for (int n = threadIdx.x; n < NPTS; n += 256) {
    float dx = xb[n] - gx;
    float d  = dx * dx;
    a0 += __expf(k0 * d);
    a1 += yb[n] * __expf(k1 * d);
  }
  s0[threadIdx.x] = a0;
  s1[threadIdx.x] = a1;
  __syncthreads();
  for (int off = 128; off > 0; off >>= 1) {
    if (threadIdx.x < (unsigned)off) {
      s0[threadIdx.x] += s0[threadIdx.x + off];
      s1[threadIdx.x] += s1[threadIdx.x + off];
    }
    __syncthreads();
  }
  if (threadIdx.x == 0) {
    float F0 = s0[0];
    float F1 = s1[0] / (F0 + 1e-8f);
    float hp[RHO_C];
#pragma unroll
    for (int j = 0; j < RHO_C; ++j) {
      float t = F0 * enc_W[j] + F1 * enc_W[RHO_C + j] + enc_b[j];
      hp[j] = 1.0f / (1.0f + __expf(-t));
    }
#pragma unroll
    for (int c = 0; c < RHO_C; ++c) {
      float acc = rho_b[c];
#pragma unroll
      for (int j = 0; j < RHO_C; ++j) acc += hp[j] * rho_W[j * RHO_C + c];
      h_out[(size_t)(b * NGRID + g) * RHO_C + c] = acc;
    }
  }
}

// ------------------------------------------------------------------
// Kernel 2: per output point m: RBF-project h over the grid ->
// mean (written directly into d_out) and basis[b,m,0:1024] (workspace).
// One wave32 per (b,m); butterfly shfl_xor reduction keeps all lanes
// holding the full sums for the coalesced 1024-wide sig_W matmul.
// ------------------------------------------------------------------
__global__ __launch_bounds__(256)
void knp_project(const float* __restrict__ x_out, const float* __restrict__ h,
                 const float* __restrict__ mean_sigma, const float* __restrict__ mean_W,
                 const float* __restrict__ mean_b, const float* __restrict__ sig_sigma,
                 const float* __restrict__ sig_W, const float* __restrict__ sig_b,
                 float* __restrict__ mean_out, float* __restrict__ basis) {
  const int lane = threadIdx.x & 31;
  const int wave = threadIdx.x >> 5;
  const int wm   = blockIdx.x * 8 + wave;   // 0 .. BATCH*MPTS-1
  const int b    = wm / MPTS;
  const int m    = wm % MPTS;
  const float xo = x_out[b * MPTS + m];

  float km[RHO_C], ks[RHO_C];
#pragma unroll
  for (int c = 0; c < RHO_C; ++c) {
    km[c] = -0.5f * __expf(-2.0f * mean_sigma[c]);
    ks[c] = -0.5f * __expf(-2.0f * sig_sigma[c]);
  }
  float mf[RHO_C], sf[RHO_C];
#pragma unroll
  for (int c = 0; c < RHO_C; ++c) { mf[c] = 0.0f; sf[c] = 0.0f; }

  const float* hb = h + (size_t)b * NGRID * RHO_C;
  for (int g = lane; g < NGRID; g += 32) {
    float gx = XMIN_F + (float)g * GRID_STEP;
    float dx = gx - xo;
    float d  = dx * dx;
    const float4* hp = (const float4*)(hb + g * RHO_C);
    float4 h0 = hp[0], h1 = hp[1];
    float hv[RHO_C] = {h0.x, h0.y, h0.z, h0.w, h1.x, h1.y, h1.z, h1.w};
#pragma unroll
    for (int c = 0; c < RHO_C; ++c) {
      mf[c] += hv[c] * __expf(km[c] * d);
      sf[c] += hv[c] * __expf(ks[c] * d);
    }
  }
  // wave32 butterfly reduction: every lane ends with the full sums
#pragma unroll
  for (int off = 16; off >= 1; off >>= 1) {
#pragma unroll
    for (int c = 0; c < RHO_C; ++c) {
      mf[c] += __shfl_xor(mf[c], off, 32);
      sf[c] += __shfl_xor(sf[c], off, 32);
    }
  }
  if (lane == 0) {
    float acc = mean_b[0];
#pragma unroll
    for (int c = 0; c < RHO_C; ++c) acc += mf[c] * mean_W[c];
    mean_out[b * MPTS + m] = acc;
  }
  float* bp = basis + (size_t)(b * MPTS + m) * SIGMA_CH;
  for (int j = lane; j < SIGMA_CH; j += 32) {
    float acc = sig_b[j];
#pragma unroll
    for (int c = 0; c < RHO_C; ++c) acc += sf[c] * sig_W[c * SIGMA_CH + j];
    bp[j] = acc;
  }
}

// ------------------------------------------------------------------
// Kernel 3: cov[b] = basis[b] * basis[b]^T + exp(noise)*I via
// V_WMMA_F32_16X16X4_F32. One wave computes a 64x64 tile (4x4 WMMA
// fragments): per 4-wide K step, 8 x b64 loads feed 16 WMMAs (2:1
// wmma:vmem), halving L2 traffic vs a 2x2 tile. The four A (and B)
// fragment rows are 64KB apart -> folded into the 24-bit instruction
// offset, so only two base addresses advance through the K loop.
// A 16x4 f32 layout: lane l holds row (l&15), K pair kb=(l<16?0:2);
// B 4x16 mirrors it with N along lanes (cov = S*S^T, so B fragments
// are just other row-blocks of basis).
// ------------------------------------------------------------------
__global__ __launch_bounds__(256)
void knp_cov_wmma(const float* __restrict__ basis, const float* __restrict__ noise_scale,
                  float* __restrict__ cov) {
  const int TILES = MPTS / 64;                       // 32 tiles per dim
  const int wgid  = blockIdx.x * 8 + (threadIdx.x >> 5);
  const int b     = wgid / (TILES * TILES);
  const int t     = wgid % (TILES * TILES);
  const int tr    = t / TILES;
  const int tc    = t % TILES;
  const int lane  = threadIdx.x & 31;
  const int mrow  = lane & 15;
  const int kb    = (lane >> 4) << 1;                // 0 or 2

  const float* Sb = basis + (size_t)b * MPTS * SIGMA_CH;
  const float* ap = Sb + (size_t)(tr * 64 + mrow) * SIGMA_CH + kb;
  const float* bp = Sb + (size_t)(tc * 64 + mrow) * SIGMA_CH + kb;

  v8f acc[4][4];
#pragma unroll
  for (int i = 0; i < 4; ++i)
#pragma unroll
    for (int j = 0; j < 4; ++j) acc[i][j] = (v8f){};

  for (int k = 0; k < SIGMA_CH; k += 4) {
    v2f af[4], bf[4];
#pragma unroll
    for (int i = 0; i < 4; ++i) {
      af[i] = *(const v2f*)(ap + (size_t)i * 16 * SIGMA_CH + k);
      bf[i] = *(const v2f*)(bp + (size_t)i * 16 * SIGMA_CH + k);
    }
#pragma unroll
    for (int i = 0; i < 4; ++i) {
#pragma unroll
      for (int j = 0; j < 4; ++j) {
        acc[i][j] = __builtin_amdgcn_wmma_f32_16x16x4_f32(
            false, af[i], false, bf[j], (short)0, acc[i][j], false, false);
      }
    }
  }

  const float noise = __expf(noise_scale[0]);
  const int n   = lane & 15;
  const int mhi = (lane >> 4) << 3;                  // 0 or 8
  float* Cb = cov + (size_t)b * MPTS * MPTS;
#pragma unroll
  for (int i = 0; i < 4; ++i) {
#pragma unroll
    for (int r = 0; r < 8; ++r) {
      const int gr = tr * 64 + i * 16 + r + mhi;
      const size_t rowoff = (size_t)gr * MPTS;
#pragma unroll
      for (int j = 0; j < 4; ++j) {
        const int gc = tc * 64 + j * 16 + n;
        float v = acc[i][j][r];
        if (gr == gc) v += noise;
        Cb[rowoff + gc] = v;
      }
    }
  }
}

extern "C" void kernel_launch(void* const* d_in, const int* in_sizes, int n_in,
                              void* d_out, int out_size, void* d_ws, size_t ws_size,
                              hipStream_t stream) {
  const float* x          = (const float*)d_in[0];
  const float* y          = (const float*)d_in[1];
  const float* x_out      = (const float*)d_in[2];
  const float* enc_sigma  = (const float*)d_in[3];
  const float* enc_W      = (const float*)d_in[4];
  const float* enc_b      = (const float*)d_in[5];
  const float* rho_W      = (const float*)d_in[6];
  const float* rho_b      = (const float*)d_in[7];
  const float* mean_sigma = (const float*)d_in[8];
  const float* mean_W     = (const float*)d_in[9];
  const float* mean_b     = (const float*)d_in[10];
  const float* sig_sigma  = (const float*)d_in[11];
  const float* sig_W      = (const float*)d_in[12];
  const float* sig_b      = (const float*)d_in[13];
  const float* noise      = (const float*)d_in[14];

  float* out_mean = (float*)d_out;                        // (B, M) flat
  float* out_cov  = (float*)d_out + (size_t)BATCH * MPTS; // (B, M, M) flat

  float* ws_basis = (float*)d_ws;                                          // B*M*1024 f32
  float* ws_h     = ws_basis + (size_t)BATCH * MPTS * SIGMA_CH;            // B*G*8 f32

  // 1) encode: F + tiny MLP -> h
  knp_encode<<<BATCH * NGRID, 256, 0, stream>>>(x, y, enc_sigma, enc_W, enc_b,
                                                rho_W, rho_b, ws_h);
  // 2) project: mean + basis
  knp_project<<<(BATCH * MPTS) / 8, 256, 0, stream>>>(x_out, ws_h,
                                                      mean_sigma, mean_W, mean_b,
                                                      sig_sigma, sig_W, sig_b,
                                                      out_mean, ws_basis);
  // 3) cov = basis * basis^T + exp(noise)*I  (fp32 WMMA, 64x64/wave)
  const int tiles = MPTS / 64;                        // 32
  const int waves = BATCH * tiles * tiles;            // 8192
  knp_cov_wmma<<<waves / 8, 256, 0, stream>>>(ws_basis, noise, out_cov);
}